// GraphConvSparse_59081570123779
// MI455X (gfx1250) — compile-verified
//
#include <hip/hip_runtime.h>
#include <hip/hip_bf16.h>
#include <stdint.h>

#define N_NODES 100000
#define NUM_EDGES 1600000
#define IN_DIM 256
#define OUT_DIM 64

typedef __attribute__((ext_vector_type(2))) float v2f;
typedef __attribute__((ext_vector_type(4))) float v4f;
typedef __attribute__((ext_vector_type(8))) float v8f;

// ---------------------------------------------------------------------------
// Kernel 1: x[N,64] = inputs[N,256] @ weight[256,64], fp32 via V_WMMA_F32_16X16X4_F32.
// 256 threads = 8 waves; each wave computes a 32-row x 64-col block (two 16-row
// strips sharing the B fragments -> 8 WMMA per 4 LDS b64 loads).
//
// Weight staged in LDS pair-interleaved:
//   wlds[p*128 + n*2 + parity] = W[(2p+parity)*64 + n]   (p = k>>1)
// so a B fragment (W[kk][n], W[kk+1][n]) is one aligned ds_load_b64 -> no
// repacking movs in front of the WMMAs; the 4 n-tiles sit at +128B steps and
// merge into ds_load_2addr_b64 pairs.
// ---------------------------------------------------------------------------
__global__ __launch_bounds__(256) void gemm_wmma_f32(const float* __restrict__ A,
                                                     const float* __restrict__ W,
                                                     float* __restrict__ X) {
  __shared__ __align__(16) float wlds[IN_DIM * OUT_DIM];  // 64 KB, 128 pair-rows x 128 floats

  const int tid = threadIdx.x;
  // Stage: 8192 pair-slots; thread handles slot i = j*256+tid.
  // Reads coalesced (consecutive lanes -> consecutive n), writes hit all 64
  // banks exactly once per wave (b64 at word 2n within a 128-word row).
#pragma unroll 4
  for (int j = 0; j < 32; ++j) {
    const int i = j * 256 + tid;  // 0..8191
    const int p = i >> 6;         // k-pair index 0..127
    const int n = i & 63;         // output column
    v2f w;
    w.x = W[(2 * p) * OUT_DIM + n];
    w.y = W[(2 * p + 1) * OUT_DIM + n];
    *(v2f*)(wlds + p * 128 + n * 2) = w;
  }
  __syncthreads();

  const int wave = tid >> 5;
  const int lane = tid & 31;
  const int hl = lane & 15;  // position within half-wave
  const int hi = lane >> 4;  // 0: lanes 0-15 (K=k,k+1), 1: lanes 16-31 (K=k+2,k+3)

  const int strip0 = blockIdx.x * 256 + wave * 32;
  int r0 = strip0 + hl;
  int r1 = strip0 + 16 + hl;
  int r0c = r0 < N_NODES ? r0 : (N_NODES - 1);  // clamp: keep EXEC all-ones for WMMA
  int r1c = r1 < N_NODES ? r1 : (N_NODES - 1);
  const float* __restrict__ a0p = A + (size_t)r0c * IN_DIM;
  const float* __restrict__ a1p = A + (size_t)r1c * IN_DIM;

  v8f acc00 = {}, acc01 = {}, acc02 = {}, acc03 = {};
  v8f acc10 = {}, acc11 = {}, acc12 = {}, acc13 = {};

#pragma unroll 4
  for (int k = 0; k < IN_DIM; k += 4) {
    const int kk = k + (hi << 1);  // this lane's K pair start
    v2f a0 = *(const v2f*)(a0p + kk);
    v2f a1 = *(const v2f*)(a1p + kk);

    const float* __restrict__ bp = wlds + ((k >> 1) + hi) * 128 + hl * 2;
    v2f b0 = *(const v2f*)(bp);       // n-tile 0 (cols  0-15)
    v2f b1 = *(const v2f*)(bp + 32);  // n-tile 1 (cols 16-31)
    v2f b2 = *(const v2f*)(bp + 64);  // n-tile 2 (cols 32-47)
    v2f b3 = *(const v2f*)(bp + 96);  // n-tile 3 (cols 48-63)

    acc00 = __builtin_amdgcn_wmma_f32_16x16x4_f32(false, a0, false, b0, (short)0, acc00, false, false);
    acc01 = __builtin_amdgcn_wmma_f32_16x16x4_f32(false, a0, false, b1, (short)0, acc01, false, false);
    acc02 = __builtin_amdgcn_wmma_f32_16x16x4_f32(false, a0, false, b2, (short)0, acc02, false, false);
    acc03 = __builtin_amdgcn_wmma_f32_16x16x4_f32(false, a0, false, b3, (short)0, acc03, false, false);
    acc10 = __builtin_amdgcn_wmma_f32_16x16x4_f32(false, a1, false, b0, (short)0, acc10, false, false);
    acc11 = __builtin_amdgcn_wmma_f32_16x16x4_f32(false, a1, false, b1, (short)0, acc11, false, false);
    acc12 = __builtin_amdgcn_wmma_f32_16x16x4_f32(false, a1, false, b2, (short)0, acc12, false, false);
    acc13 = __builtin_amdgcn_wmma_f32_16x16x4_f32(false, a1, false, b3, (short)0, acc13, false, false);
  }

  // C/D layout: VGPR v, lanes 0-15 -> M=v, N=lane; lanes 16-31 -> M=v+8, N=lane-16
  const int baseR0 = strip0 + hi * 8;
  const int baseR1 = strip0 + 16 + hi * 8;
#pragma unroll
  for (int v = 0; v < 8; ++v) {
    const int ra = baseR0 + v;
    if (ra < N_NODES) {
      float* xr = X + (size_t)ra * OUT_DIM + hl;
      xr[0]  = acc00[v];
      xr[16] = acc01[v];
      xr[32] = acc02[v];
      xr[48] = acc03[v];
    }
    const int rb = baseR1 + v;
    if (rb < N_NODES) {
      float* xr = X + (size_t)rb * OUT_DIM + hl;
      xr[0]  = acc10[v];
      xr[16] = acc11[v];
      xr[32] = acc12[v];
      xr[48] = acc13[v];
    }
  }
}

// ---------------------------------------------------------------------------
// Kernel 2: agg[r] += x[c] per edge. 16 lanes cooperate per edge (4 floats each).
// x (25.6 MB) and agg (25.6 MB) both fit in the 192 MB L2 -> gathers and the
// native f32 atomics resolve at L2, not HBM.
// ---------------------------------------------------------------------------
__global__ __launch_bounds__(256) void scatter_atomic(const float* __restrict__ X,
                                                      const long long* __restrict__ rows,
                                                      const long long* __restrict__ cols,
                                                      float* __restrict__ out,
                                                      int nE) {
  const int t = blockIdx.x * 256 + threadIdx.x;
  const int e = t >> 4;
  const int s = t & 15;
  if (e >= nE) return;
  const int r = (int)rows[e];
  const int c = (int)cols[e];
  const v4f v = *(const v4f*)(X + (size_t)c * OUT_DIM + s * 4);
  float* dst = out + (size_t)r * OUT_DIM + s * 4;
  unsafeAtomicAdd(dst + 0, v.x);
  unsafeAtomicAdd(dst + 1, v.y);
  unsafeAtomicAdd(dst + 2, v.z);
  unsafeAtomicAdd(dst + 3, v.w);
}

// ---------------------------------------------------------------------------
// Kernel 3: in-place sigmoid, float4-vectorized (out_size = 6.4M, divisible by 4).
// ---------------------------------------------------------------------------
__global__ __launch_bounds__(256) void sigmoid_k(float* __restrict__ out, int n4) {
  const int i = blockIdx.x * 256 + threadIdx.x;
  if (i >= n4) return;
  v4f v = *((v4f*)out + i);
  v.x = 1.0f / (1.0f + __expf(-v.x));
  v.y = 1.0f / (1.0f + __expf(-v.y));
  v.z = 1.0f / (1.0f + __expf(-v.z));
  v.w = 1.0f / (1.0f + __expf(-v.w));
  *((v4f*)out + i) = v;
}

extern "C" void kernel_launch(void* const* d_in, const int* in_sizes, int n_in,
                              void* d_out, int out_size, void* d_ws, size_t ws_size,
                              hipStream_t stream) {
  const float*     inputs = (const float*)d_in[0];
  const long long* edges  = (const long long*)d_in[1];  // int64: [2, N_EDGES]
  const float*     weight = (const float*)d_in[2];
  float* out = (float*)d_out;
  float* x   = (float*)d_ws;  // N_NODES * OUT_DIM fp32 = 25.6 MB scratch

  // 1) Dense projection (WMMA fp32): 256 rows per block
  gemm_wmma_f32<<<(N_NODES + 255) / 256, 256, 0, stream>>>(inputs, weight, x);

  // 2) Zero accumulator, then edge scatter-add
  hipMemsetAsync(d_out, 0, (size_t)out_size * sizeof(float), stream);
  const long long nEdges = in_sizes[1] / 2;  // 1.6M
  scatter_atomic<<<(unsigned)((nEdges * 16 + 255) / 256), 256, 0, stream>>>(
      x, edges, edges + nEdges, out, (int)nEdges);

  // 3) Sigmoid in place
  const int n4 = out_size / 4;
  sigmoid_k<<<(n4 + 255) / 256, 256, 0, stream>>>(out, n4);
}